// RelationAwareGNN_45595372814351
// MI455X (gfx1250) — compile-verified
//
#include <hip/hip_runtime.h>

#define HIDDEN 128
#define INDIM  256
#define LN_EPS 1e-5f
#define LSTR    130   // padded LDS row stride (floats) for 128-wide weight rows
#define LSTR_IN 258   // padded LDS row stride (floats) for 256-wide weight rows

typedef float v2f __attribute__((ext_vector_type(2)));
typedef float v8f __attribute__((ext_vector_type(8)));

// D = A(16x4 f32) * B(4x16 f32) + C(16x16 f32), wave32 WMMA.
__device__ __forceinline__ v8f wmma4(v2f a, v2f b, v8f c) {
  return __builtin_amdgcn_wmma_f32_16x16x4_f32(
      /*neg_a=*/false, a, /*neg_b=*/false, b,
      /*c_mod=*/(short)0, c, /*reuse_a=*/false, /*reuse_b=*/false);
}

__global__ void zero_f32(float* __restrict__ p, size_t n) {
  size_t i = (size_t)blockIdx.x * blockDim.x + threadIdx.x;
  size_t st = (size_t)gridDim.x * blockDim.x;
  for (; i < n; i += st) p[i] = 0.0f;
}

__global__ void deg_accum(const int* __restrict__ dst, float* __restrict__ deg, int E) {
  int e = blockIdx.x * blockDim.x + threadIdx.x;
  if (e < E) atomicAdd(&deg[dst[e]], 1.0f);
}

__global__ void deg_finalize(const float* __restrict__ deg, float* __restrict__ invd,
                             float* __restrict__ mask, int N) {
  int i = blockIdx.x * blockDim.x + threadIdx.x;
  if (i < N) {
    float d = deg[i];
    bool has = d > 0.0f;
    invd[i] = has ? 1.0f / d : 0.0f;
    mask[i] = has ? 1.0f : 0.0f;
  }
}

// One 32-lane group per edge; each lane handles 4 contiguous channels.
__global__ void edge_scatter(const float* __restrict__ h, const float* __restrict__ rel_emb,
                             const int* __restrict__ src, const int* __restrict__ dst,
                             const int* __restrict__ rel, float* __restrict__ agg, int E) {
  int gid = blockIdx.x * blockDim.x + threadIdx.x;
  int e = gid >> 5;
  if (e >= E) return;
  int lane = gid & 31;
  int c = lane << 2;
  int s = src[e], d = dst[e], r = rel[e];
  const float4 hv = *(const float4*)(h + (size_t)s * HIDDEN + c);
  const float4 rv = *(const float4*)(rel_emb + (size_t)r * HIDDEN + c);
  float* out = agg + (size_t)d * HIDDEN + c;
  atomicAdd(out + 0, hv.x + rv.x);
  atomicAdd(out + 1, hv.y + rv.y);
  atomicAdd(out + 2, hv.z + rv.z);
  atomicAdd(out + 3, hv.w + rv.w);
}

// h0 = x @ W_in^T + b_in. 256 threads = 8 waves = 128 rows/block.
// W_in staged once into LDS (padded stride) and shared by all 8 waves.
__global__ void __launch_bounds__(256) input_proj(const float* __restrict__ x,
                                                  const float* __restrict__ W,   // [128,256]
                                                  const float* __restrict__ bias,
                                                  float* __restrict__ out, int N) {
  extern __shared__ float smem[];   // 128 * LSTR_IN floats
  const int tid = threadIdx.x;

  for (int i = tid; i < HIDDEN * INDIM / 4; i += 256) {
    const int r = (i * 4) >> 8;           // /INDIM
    const int c = (i * 4) & (INDIM - 1);
    *(float4*)&smem[r * LSTR_IN + c] = *(const float4*)&W[(size_t)r * INDIM + c];
  }
  __syncthreads();

  const int lane = tid & 31;
  const int wv   = tid >> 5;
  const int base = blockIdx.x * 128 + wv * 16;
  const int m    = lane & 15;     // A row / B,C column within tile
  const int half = lane >> 4;     // selects K sub-pair / row+8
  int arow = base + m; if (arow >= N) arow = N - 1;
  const float* xrow = x + (size_t)arow * INDIM;

  v8f zero = {};
  v8f acc[8];
#pragma unroll
  for (int t = 0; t < 8; ++t) acc[t] = zero;

  for (int kb = 0; kb < INDIM; kb += 4) {
    const int k0 = kb + half * 2;
    v2f a = *(const v2f*)(xrow + k0);
#pragma unroll
    for (int t = 0; t < 8; ++t) {
      const int n = t * 16 + m;
      v2f b = *(const v2f*)&smem[n * LSTR_IN + k0];
      acc[t] = wmma4(a, b, acc[t]);
    }
  }

#pragma unroll
  for (int t = 0; t < 8; ++t) {
    const int n = t * 16 + m;
    const float bv = bias[n];
#pragma unroll
    for (int v = 0; v < 8; ++v) {
      const int r = base + v + half * 8;
      if (r < N) out[(size_t)r * HIDDEN + n] = acc[t][v] + bv;
    }
  }
}

// Fused per-layer: t = h@W_self^T + (agg*inv_deg)@W_neigh^T + b_self + mask*b_neigh
//                  h_out = relu(layernorm(t) * g + b)
// 256 threads = 8 waves = 128 rows/block; both weight matrices staged in LDS.
__global__ void __launch_bounds__(256) layer_fused(
    const float* __restrict__ h_in, const float* __restrict__ agg,
    const float* __restrict__ invd, const float* __restrict__ mask,
    const float* __restrict__ Wn, const float* __restrict__ bn,
    const float* __restrict__ Ws, const float* __restrict__ bs,
    const float* __restrict__ lng, const float* __restrict__ lnb,
    float* __restrict__ h_out, int N) {
  extern __shared__ float smem[];           // 2 * 128 * LSTR floats
  float* WsL = smem;
  float* WnL = smem + HIDDEN * LSTR;
  const int tid = threadIdx.x;

  for (int i = tid; i < HIDDEN * HIDDEN / 4; i += 256) {
    const int r = (i * 4) >> 7;             // /HIDDEN
    const int c = (i * 4) & (HIDDEN - 1);
    *(float4*)&WsL[r * LSTR + c] = *(const float4*)&Ws[(size_t)r * HIDDEN + c];
    *(float4*)&WnL[r * LSTR + c] = *(const float4*)&Wn[(size_t)r * HIDDEN + c];
  }
  __syncthreads();

  const int lane = tid & 31;
  const int wv   = tid >> 5;
  const int base = blockIdx.x * 128 + wv * 16;
  const int m    = lane & 15;
  const int half = lane >> 4;
  int arow = base + m; if (arow >= N) arow = N - 1;
  const float idg = invd[arow];
  const float* hrow = h_in + (size_t)arow * HIDDEN;
  const float* grow = agg + (size_t)arow * HIDDEN;

  v8f zero = {};
  v8f acc[8];
#pragma unroll
  for (int t = 0; t < 8; ++t) acc[t] = zero;

  for (int kb = 0; kb < HIDDEN; kb += 4) {
    const int k0 = kb + half * 2;
    v2f a_s = *(const v2f*)(hrow + k0);
    v2f a_n = *(const v2f*)(grow + k0);
    a_n.x *= idg; a_n.y *= idg;
#pragma unroll
    for (int t = 0; t < 8; ++t) {
      const int n = t * 16 + m;
      v2f b_s = *(const v2f*)&WsL[n * LSTR + k0];
      v2f b_n = *(const v2f*)&WnL[n * LSTR + k0];
      acc[t] = wmma4(a_s, b_s, acc[t]);
      acc[t] = wmma4(a_n, b_n, acc[t]);
    }
  }

  // bias + neighbor mask (row-dependent mask lives on the VGPR index axis)
  float mv[8];
#pragma unroll
  for (int v = 0; v < 8; ++v) {
    int r = base + v + half * 8; if (r >= N) r = N - 1;
    mv[v] = mask[r];
  }
#pragma unroll
  for (int t = 0; t < 8; ++t) {
    const int n = t * 16 + m;
    const float bsv = bs[n], bnv = bn[n];
#pragma unroll
    for (int v = 0; v < 8; ++v) acc[t][v] += bsv + mv[v] * bnv;
  }

  // LayerNorm: rows = VGPR index (per 16-lane half), cols = lanes x tiles.
  // xor masks 1..8 stay inside each 16-lane half of the wave32.
  v8f rsum = zero;
#pragma unroll
  for (int t = 0; t < 8; ++t) rsum += acc[t];
#pragma unroll
  for (int off = 1; off < 16; off <<= 1) {
#pragma unroll
    for (int v = 0; v < 8; ++v) rsum[v] += __shfl_xor(rsum[v], off, 32);
  }
  v8f mean;
#pragma unroll
  for (int v = 0; v < 8; ++v) mean[v] = rsum[v] * (1.0f / HIDDEN);

  v8f dsum = zero;
#pragma unroll
  for (int t = 0; t < 8; ++t) {
    v8f d = acc[t] - mean;
    dsum += d * d;
  }
#pragma unroll
  for (int off = 1; off < 16; off <<= 1) {
#pragma unroll
    for (int v = 0; v < 8; ++v) dsum[v] += __shfl_xor(dsum[v], off, 32);
  }
  float rstd[8];
#pragma unroll
  for (int v = 0; v < 8; ++v) rstd[v] = rsqrtf(dsum[v] * (1.0f / HIDDEN) + LN_EPS);

#pragma unroll
  for (int t = 0; t < 8; ++t) {
    const int n = t * 16 + m;
    const float gv = lng[n], bv = lnb[n];
#pragma unroll
    for (int v = 0; v < 8; ++v) {
      const int r = base + v + half * 8;
      float val = (acc[t][v] - mean[v]) * rstd[v] * gv + bv;
      val = fmaxf(val, 0.0f);
      if (r < N) h_out[(size_t)r * HIDDEN + n] = val;
    }
  }
}

extern "C" void kernel_launch(void* const* d_in, const int* in_sizes, int n_in,
                              void* d_out, int out_size, void* d_ws, size_t ws_size,
                              hipStream_t stream) {
  const float* node_feat = (const float*)d_in[0];
  const int*   src       = (const int*)d_in[1];
  const int*   dst       = (const int*)d_in[2];
  const int*   rel       = (const int*)d_in[3];
  const float* W_in      = (const float*)d_in[4];
  const float* b_in      = (const float*)d_in[5];
  const float* rel_emb   = (const float*)d_in[6];
  const float* W_neigh   = (const float*)d_in[7];   // [2,128,128]
  const float* b_neigh   = (const float*)d_in[8];   // [2,128]
  const float* W_self    = (const float*)d_in[9];   // [2,128,128]
  const float* b_self    = (const float*)d_in[10];  // [2,128]
  const float* ln_g      = (const float*)d_in[11];  // [2,128]
  const float* ln_b      = (const float*)d_in[12];  // [2,128]

  const int N = in_sizes[0] / INDIM;
  const int E = in_sizes[1];

  float* ws   = (float*)d_ws;
  float* h0   = ws;                          // N*128
  float* agg  = h0  + (size_t)N * HIDDEN;    // N*128
  float* deg  = agg + (size_t)N * HIDDEN;    // N
  float* invd = deg + N;                     // N
  float* mask = invd + N;                    // N
  float* hout = (float*)d_out;

  zero_f32<<<512, 256, 0, stream>>>(deg, (size_t)N);
  deg_accum<<<(E + 255) / 256, 256, 0, stream>>>(dst, deg, E);
  deg_finalize<<<(N + 255) / 256, 256, 0, stream>>>(deg, invd, mask, N);

  const int blocks128 = (N + 127) / 128;
  const size_t smem_in    = (size_t)HIDDEN * LSTR_IN * sizeof(float);   // ~129 KB
  const size_t smem_layer = (size_t)2 * HIDDEN * LSTR * sizeof(float);  // ~130 KB

  input_proj<<<blocks128, 256, smem_in, stream>>>(node_feat, W_in, b_in, h0, N);

  const float* hcur = h0;
  for (int l = 0; l < 2; ++l) {
    zero_f32<<<2048, 256, 0, stream>>>(agg, (size_t)N * HIDDEN);
    edge_scatter<<<(int)(((size_t)E * 32 + 255) / 256), 256, 0, stream>>>(
        hcur, rel_emb, src, dst, rel, agg, E);
    layer_fused<<<blocks128, 256, smem_layer, stream>>>(
        hcur, agg, invd, mask,
        W_neigh + (size_t)l * HIDDEN * HIDDEN, b_neigh + (size_t)l * HIDDEN,
        W_self  + (size_t)l * HIDDEN * HIDDEN, b_self  + (size_t)l * HIDDEN,
        ln_g + (size_t)l * HIDDEN, ln_b + (size_t)l * HIDDEN,
        hout, N);
    hcur = hout;  // layer 2 runs in-place on d_out (block reads only rows it writes)
  }
}